// AttentionPooling_V_58849641890561
// MI455X (gfx1250) — compile-verified
//
#include <hip/hip_runtime.h>
#include <hip/hip_bf16.h>
#include <math.h>

typedef __attribute__((ext_vector_type(2))) float v2f;
typedef __attribute__((ext_vector_type(8))) float v8f;

#define B_  4
#define N_  16384
#define E_  128
#define H_  64
#define NS_ 2048
#define K_  27

// Flat output layout (elements; indices stored as int32 bit patterns)
#define OUT_XOUT 0
#define OUT_XVN  (B_*NS_*E_)
#define OUT_XS   (OUT_XVN + B_*NS_*3)
#define OUT_PIDX (OUT_XS + B_*N_)
#define OUT_UIDX (OUT_PIDX + B_*NS_*K_)

// ---------------------------------------------------------------------------
// CDNA5 async global->LDS copy (ASYNCcnt-tracked), inline-asm (portable per
// bridge doc). LDS dest operand = low 32 bits of generic shared pointer
// (= addrspace(3) byte offset).
// ---------------------------------------------------------------------------
__device__ __forceinline__ void async_copy_b128(const void* gsrc, void* lds_dst) {
  unsigned l = (unsigned)(unsigned long long)(uintptr_t)lds_dst;
  asm volatile("global_load_async_to_lds_b128 %0, %1, off"
               :: "v"(l), "v"(gsrc) : "memory");
}
__device__ __forceinline__ void wait_async_le1() {
  asm volatile("s_wait_asynccnt 0x1" ::: "memory");
}
__device__ __forceinline__ void wait_async_0() {
  asm volatile("s_wait_asynccnt 0x0" ::: "memory");
}

// ---------------------------------------------------------------------------
// Pack every x_v point as (x, y, z, |p|^2) -> one b128 per point in KNN
// ---------------------------------------------------------------------------
__global__ void pack_kernel(const float* __restrict__ xv,
                            float4* __restrict__ packF) {
  int i = blockIdx.x * blockDim.x + threadIdx.x;      // over B*N
  const float* p = xv + (size_t)i * 3;
  float x = p[0], y = p[1], z = p[2];
  packF[i] = make_float4(x, y, z, x*x + y*y + z*z);
}

// ---------------------------------------------------------------------------
// Gather sampled points: write x_v_next output + packed (x,y,z,norm)
// ---------------------------------------------------------------------------
__global__ void sample_kernel(const float* __restrict__ xv,
                              const int* __restrict__ sidx,
                              float* __restrict__ xvn,
                              float4* __restrict__ packS) {
  int i = blockIdx.x * blockDim.x + threadIdx.x;      // over B*NS
  int b = i / NS_;
  int s = sidx[i];
  const float* p = xv + ((size_t)b * N_ + s) * 3;
  float x = p[0], y = p[1], z = p[2];
  xvn[(size_t)i*3 + 0] = x;
  xvn[(size_t)i*3 + 1] = y;
  xvn[(size_t)i*3 + 2] = z;
  packS[i] = make_float4(x, y, z, x*x + y*y + z*z);
}

// ---------------------------------------------------------------------------
// MLP score: x_s = sigmoid( tanh(x @ W1^T + b1) @ V^T + Vb )
// One wave per 16 points, WMMA f32 16x16x4, K-loop over E=128 (32 steps),
// 4 N-tiles covering H=64. C layout: lane L -> col h=(L%16), rows r+8*(L>=16).
// ---------------------------------------------------------------------------
__global__ void __launch_bounds__(256) score_kernel(
    const float* __restrict__ x,  const float* __restrict__ W1,
    const float* __restrict__ b1, const float* __restrict__ Vw,
    const float* __restrict__ Vb, float* __restrict__ xs) {
  int lane = threadIdx.x & 31;
  int wave = threadIdx.x >> 5;
  int tile = blockIdx.x * 8 + wave;                   // 16-point tile
  int lh   = lane & 15;
  bool hi  = lane >= 16;

  v8f acc0 = {}, acc1 = {}, acc2 = {}, acc3 = {};
  const float* xr = x + (size_t)(tile * 16 + lh) * E_ + (hi ? 2 : 0);
  const float* w0 = W1 + (size_t)lh * E_ + (hi ? 2 : 0);   // h = lh (+16*ht)

  for (int e0 = 0; e0 < E_; e0 += 4) {
    v2f a;  a.x = xr[e0];  a.y = xr[e0 + 1];
    v2f bm;
    bm.x = w0[e0];             bm.y = w0[e0 + 1];
    acc0 = __builtin_amdgcn_wmma_f32_16x16x4_f32(false, a, false, bm, (short)0, acc0, false, false);
    bm.x = w0[16*E_ + e0];     bm.y = w0[16*E_ + e0 + 1];
    acc1 = __builtin_amdgcn_wmma_f32_16x16x4_f32(false, a, false, bm, (short)0, acc1, false, false);
    bm.x = w0[32*E_ + e0];     bm.y = w0[32*E_ + e0 + 1];
    acc2 = __builtin_amdgcn_wmma_f32_16x16x4_f32(false, a, false, bm, (short)0, acc2, false, false);
    bm.x = w0[48*E_ + e0];     bm.y = w0[48*E_ + e0 + 1];
    acc3 = __builtin_amdgcn_wmma_f32_16x16x4_f32(false, a, false, bm, (short)0, acc3, false, false);
  }

  float vb = Vb[0];
  float b1v0 = b1[lh],      b1v1 = b1[16 + lh];
  float b1v2 = b1[32 + lh], b1v3 = b1[48 + lh];
  float vw0 = Vw[lh],       vw1 = Vw[16 + lh];
  float vw2 = Vw[32 + lh],  vw3 = Vw[48 + lh];

  float part[8];
#pragma unroll
  for (int r = 0; r < 8; ++r) {
    part[r] = tanhf(acc0[r] + b1v0) * vw0
            + tanhf(acc1[r] + b1v1) * vw1
            + tanhf(acc2[r] + b1v2) * vw2
            + tanhf(acc3[r] + b1v3) * vw3;
  }
#pragma unroll
  for (int m = 1; m < 16; m <<= 1) {
#pragma unroll
    for (int r = 0; r < 8; ++r) part[r] += __shfl_xor(part[r], m, 32);
  }
  if (lh == 0) {                                       // lanes 0 and 16
#pragma unroll
    for (int r = 0; r < 8; ++r) {
      int row = tile * 16 + r + (hi ? 8 : 0);
      float v = part[r] + vb;
      xs[row] = 1.0f / (1.0f + expf(-v));
    }
  }
}

// ---------------------------------------------------------------------------
// Brute-force KNN, K=27. Distance tile via WMMA f32 16x16x4:
//   A row m (ref):   ( rx, ry, rz, |r|^2 )    (packed float4 from LDS)
//   B col n (query): (-2qx,-2qy,-2qz, 1 )
//   C[m][n] = |r|^2 - 2 r.q   (|q|^2 dropped: per-column constant, ordering OK)
// Refs are streamed in 128-point chunks via async global->LDS (double buffer,
// ASYNCcnt). Each lane owns one query column; per-lane sorted top-27 in
// registers; 8 sorted lists per query merged in LDS.
// One workgroup = 16 queries, 4 waves; wave w computes tiles 2w,2w+1 / chunk.
// ---------------------------------------------------------------------------
__global__ void __launch_bounds__(128) knn_kernel(
    const float4* __restrict__ refs, const float* __restrict__ queries,
    int* __restrict__ outIdx, int R, int Q) {
  __shared__ float4 sbuf[2][128];
  __shared__ float ldsd[4][32][K_];
  __shared__ int   ldsi[4][32][K_];

  int qtpb = Q >> 4;
  int b  = blockIdx.x / qtpb;
  int qt = blockIdx.x % qtpb;
  const float4* refb = refs    + (size_t)b * R;
  const float*  qb   = queries + ((size_t)b * Q + qt * 16) * 3;

  int tid  = threadIdx.x;
  int lane = tid & 31;
  int wave = tid >> 5;
  int lh   = lane & 15;
  bool hi  = lane >= 16;

  // B operand (queries), built once
  const float* qp = qb + (size_t)lh * 3;
  float qx = qp[0], qy = qp[1], qz = qp[2];
  v2f bm;
  bm.x = hi ? (-2.0f * qz) : (-2.0f * qx);
  bm.y = hi ? 1.0f         : (-2.0f * qy);

  float bd[K_]; int bi[K_];
#pragma unroll
  for (int k = 0; k < K_; ++k) { bd[k] = 3.4e38f; bi[k] = 0; }

  int nch = R >> 7;                       // 128-point chunks (R % 128 == 0)
  // async double-buffer prologue: each thread fetches one float4 per chunk
  async_copy_b128(refb + tid,        &sbuf[0][tid]);
  if (nch > 1) async_copy_b128(refb + 128 + tid, &sbuf[1][tid]);

  for (int c = 0; c < nch; ++c) {
    if (c + 1 < nch) wait_async_le1();    // chunk c landed (in-order retire)
    else             wait_async_0();
    __syncthreads();                      // publish buffer across waves
    const float4* cb = sbuf[c & 1];
#pragma unroll
    for (int tt = 0; tt < 2; ++tt) {
      int t = 2 * wave + tt;
      float4 p = cb[t * 16 + lh];
      v2f am;
      am.x = hi ? p.z : p.x;
      am.y = hi ? p.w : p.y;
      v8f cc = {};
      cc = __builtin_amdgcn_wmma_f32_16x16x4_f32(false, am, false, bm, (short)0, cc, false, false);

      int rowoff = (c << 7) + t * 16 + (hi ? 8 : 0);
#pragma unroll
      for (int r = 0; r < 8; ++r) {
        float d = cc[r];
        int   id = rowoff + r;
        if (d < bd[K_ - 1]) {             // beats current worst -> insert
          bd[K_ - 1] = d; bi[K_ - 1] = id;
#pragma unroll
          for (int j = K_ - 1; j > 0; --j) {
            if (bd[j] < bd[j - 1]) {
              float td = bd[j]; bd[j] = bd[j - 1]; bd[j - 1] = td;
              int   ti = bi[j]; bi[j] = bi[j - 1]; bi[j - 1] = ti;
            }
          }
        }
      }
    }
    __syncthreads();                      // all waves done reading buf[c&1]
    if (c + 2 < nch)
      async_copy_b128(refb + (c + 2) * 128 + tid, &sbuf[c & 1][tid]);
  }

#pragma unroll
  for (int k = 0; k < K_; ++k) {
    ldsd[wave][lane][k] = bd[k];
    ldsi[wave][lane][k] = bi[k];
  }
  __syncthreads();

  if (tid < 16) {                         // 8-way merge of sorted lists
    int pos[8] = {0,0,0,0,0,0,0,0};
    int* op = outIdx + ((size_t)b * Q + qt * 16 + tid) * K_;
    for (int k = 0; k < K_; ++k) {
      float best = 3.4e38f; int bj = 0;
#pragma unroll
      for (int j = 0; j < 8; ++j) {
        float dv = ldsd[j >> 1][(j & 1) ? (tid + 16) : tid][pos[j]];
        if (dv < best) { best = dv; bj = j; }
      }
      op[k] = ldsi[bj >> 1][(bj & 1) ? (tid + 16) : tid][pos[bj]];
      pos[bj]++;
    }
  }
}

// ---------------------------------------------------------------------------
// x_out[b,s,:] = sum_k ( score_k / sum score ) * x[b, idx_k, :]
// ---------------------------------------------------------------------------
__global__ void __launch_bounds__(128) pool_agg_kernel(
    const float* __restrict__ x, const float* __restrict__ xs,
    const int* __restrict__ pidx, float* __restrict__ xout) {
  int blk = blockIdx.x;                 // b*NS + s
  int b   = blk / NS_;
  int t   = threadIdx.x;
  __shared__ int   sidx[K_];
  __shared__ float sw[K_];
  __shared__ float ssum;
  if (t < K_) {
    int id = pidx[(size_t)blk * K_ + t];
    sidx[t] = id;
    sw[t]   = xs[(size_t)b * N_ + id];
  }
  __syncthreads();
  if (t == 0) { float s = 0.f; for (int k = 0; k < K_; ++k) s += sw[k]; ssum = s; }
  __syncthreads();
  float inv = 1.0f / ssum;
  float acc = 0.f;
  for (int k = 0; k < K_; ++k)
    acc += sw[k] * inv * x[((size_t)b * N_ + sidx[k]) * E_ + t];
  xout[(size_t)blk * E_ + t] = acc;
}

// ---------------------------------------------------------------------------
extern "C" void kernel_launch(void* const* d_in, const int* in_sizes, int n_in,
                              void* d_out, int out_size, void* d_ws, size_t ws_size,
                              hipStream_t stream) {
  const float* x    = (const float*)d_in[0];
  const float* xv   = (const float*)d_in[1];
  const float* W1w  = (const float*)d_in[2];
  const float* W1b  = (const float*)d_in[3];
  const float* Vw   = (const float*)d_in[4];
  const float* Vb   = (const float*)d_in[5];
  const int*   sidx = (const int*)d_in[6];

  float* out  = (float*)d_out;
  float* xout = out + OUT_XOUT;
  float* xvn  = out + OUT_XVN;
  float* xs   = out + OUT_XS;
  int*   pidx = (int*)(out + OUT_PIDX);
  int*   uidx = (int*)(out + OUT_UIDX);

  float4* packF = (float4*)d_ws;                       // B*N  packed refs
  float4* packS = packF + (size_t)B_ * N_;             // B*NS packed samples

  pack_kernel<<<(B_ * N_) / 256, 256, 0, stream>>>(xv, packF);
  sample_kernel<<<(B_ * NS_) / 256, 256, 0, stream>>>(xv, sidx, xvn, packS);
  score_kernel<<<(B_ * N_) / (16 * 8), 256, 0, stream>>>(x, W1w, W1b, Vw, Vb, xs);
  // pooling: queries = x_v_next (NS), refs = x_v (N) -> idx into N
  knn_kernel<<<B_ * (NS_ / 16), 128, 0, stream>>>(packF, xvn, pidx, N_, NS_);
  // unpooling: queries = x_v (N), refs = x_v_next (NS) -> idx into NS
  knn_kernel<<<B_ * (N_ / 16), 128, 0, stream>>>(packS, xv, uidx, NS_, N_);
  pool_agg_kernel<<<B_ * NS_, 128, 0, stream>>>(x, xs, pidx, xout);
}